// LiftSplatShoot_20641612825470
// MI455X (gfx1250) — compile-verified
//
#include <hip/hip_runtime.h>

// ---------------------------------------------------------------------------
// Lift-Splat-Shoot voxel pooling for MI455X (gfx1250, wave32).
//
//   out[c, vx, vy] = sum_{p, d : vox(d,p) = (vx,vy)} depth[d,p] * feat[c,p]
//
// Reformulated as:  W[v,p] = sum_d depth[d,p] [vox(d,p)==v]   (atomic build)
//                   out[v,:] = W[v,:] @ featT                  (f32 WMMA GEMM)
// over the analytically-bounded occupied voxel box [80,120)x[88,120).
//
// Roofline: output writes (82 MB) ~= 3.5us @ 23.3 TB/s dominate; the GEMM
// (M=1280,N=512,K=960 f32) is fed from LDS staged with async global->LDS
// b128 transfers (ASYNCcnt) to keep L2 fragment traffic ~120 MB.
// ---------------------------------------------------------------------------

typedef __attribute__((ext_vector_type(2))) float v2f;
typedef __attribute__((ext_vector_type(8))) float v8f;
typedef __attribute__((ext_vector_type(4))) int   v4i;

typedef __attribute__((address_space(1))) v4i* global_v4i_p;
typedef __attribute__((address_space(3))) v4i* lds_v4i_p;

#define C_CH   512
#define FH     24
#define FW     40
#define NPIX   (FH * FW)          // 960
#define ND     80
#define NPTS   (NPIX * ND)        // 76800
#define NX0    200
#define NX1    200
#define OUT_N  (C_CH * NX0 * NX1) // 20,480,000 floats

// occupied-voxel bounding box (analytic: x_vox in [81,118], y_vox in [89,112])
#define BX0    80
#define BY0    88
#define BNX    40
#define BNY    32
#define MVOX   (BNX * BNY)        // 1280

// GEMM tiling
#define KC        64              // K-chunk staged in LDS
#define A_STRIDE  68              // LDS row stride (floats): bank = (4m+k)%64
#define B_STRIDE  80              // LDS row stride (floats): halves 32 banks apart

#if __has_builtin(__builtin_amdgcn_global_load_async_to_lds_b128)
#define HAVE_ASYNC_LDS 1
#endif

__device__ __forceinline__ void async_copy_b128(const float* g, float* l) {
#if defined(HAVE_ASYNC_LDS)
    __builtin_amdgcn_global_load_async_to_lds_b128(
        (global_v4i_p)(g),       // global src  (addrspace 1, v4i pointee)
        (lds_v4i_p)(l),          // LDS dst     (addrspace 3, v4i pointee)
        /*imm offset=*/0, /*cpol=*/0);
#else
    *(float4*)l = *(const float4*)g;   // global_load_b128 + ds_store_b128
#endif
}

__device__ __forceinline__ void wait_async_copies() {
#if defined(HAVE_ASYNC_LDS)
#if __has_builtin(__builtin_amdgcn_s_wait_asynccnt)
    __builtin_amdgcn_s_wait_asynccnt(0);
#else
    asm volatile("s_wait_asynccnt 0" ::: "memory");
#endif
#endif
}

// ---------------------------------------------------------------------------
// Zero-fill (vectorized b128 stores)
// ---------------------------------------------------------------------------
__global__ __launch_bounds__(256) void lss_zero4(float4* __restrict__ p, int n4) {
    int i = blockIdx.x * blockDim.x + threadIdx.x;
    if (i < n4) p[i] = make_float4(0.f, 0.f, 0.f, 0.f);
}

// ---------------------------------------------------------------------------
// Transpose feat (C, P) -> featT (P, C); writes coalesced over C.
// ---------------------------------------------------------------------------
__global__ __launch_bounds__(256) void lss_transpose(const float* __restrict__ feat,
                                                     float* __restrict__ featT) {
    int i = blockIdx.x * blockDim.x + threadIdx.x;       // i = p*512 + c
    if (i >= NPIX * C_CH) return;
    int p = i >> 9;            // / 512
    int c = i & (C_CH - 1);
    featT[i] = feat[c * NPIX + p];
}

// ---------------------------------------------------------------------------
// Build W[v_local, p] = sum of depth over bins mapping to voxel v.
// Index math in double to mirror the numpy float64 precompute.
// ---------------------------------------------------------------------------
__global__ __launch_bounds__(256) void lss_build_w(const float* __restrict__ depth,
                                                   float* __restrict__ W) {
    int t = blockIdx.x * blockDim.x + threadIdx.x;       // t = di*NPIX + p
    if (t >= NPTS) return;
    int di = t / NPIX;
    int p  = t - di * NPIX;
    int h  = p / FW;
    int w  = p - h * FW;

    double d  = 4.0 + 0.5 * (double)di;
    double xp = (double)w * (639.0 / 39.0);              // linspace(0,639,40)
    double yp = (double)h * (383.0 / 23.0);              // linspace(0,383,24)
    double xc = d * (xp - 320.0) * (1.0 / 320.0);        // inv(K) ray
    double yc = d * (yp - 180.0) * (1.0 / 320.0);

    int vx = (int)trunc((xc + 50.0) * 2.0);
    int vy = (int)trunc((yc + 50.0) * 2.0);
    int vz = (int)trunc((d + 10.0) * (1.0 / 20.0));
    if (vx < 0 || vx >= NX0 || vy < 0 || vy >= NX1 || vz != 0) return;

    int lx = vx - BX0, ly = vy - BY0;
    if ((unsigned)lx >= BNX || (unsigned)ly >= BNY) return;   // analytically unreachable

    atomicAdd(&W[(size_t)(lx * BNY + ly) * NPIX + p], depth[t]);
}

// ---------------------------------------------------------------------------
// GEMM: out_box = W (M=1280 x K=960) @ featT (K=960 x N=512) via
// V_WMMA_F32_16X16X4_F32, A/B staged through LDS with async b128 copies.
// One wave -> one 16x16 tile, 8 waves/WG -> 32(M) x 64(N).
// ---------------------------------------------------------------------------
__global__ __launch_bounds__(256) void lss_wmma_gemm(const float* __restrict__ W,
                                                     const float* __restrict__ featT,
                                                     float* __restrict__ out) {
    __shared__ __align__(16) float As[32 * A_STRIDE];   // 8704 B
    __shared__ __align__(16) float Bs[KC * B_STRIDE];   // 20480 B

    const int tid  = threadIdx.x;
    const int lane = tid & 31;
    const int wave = tid >> 5;
    const int tm = wave >> 2;                  // 0..1
    const int tn = wave & 3;                   // 0..3
    const int Mblk = blockIdx.x * 32;          // WG row base (voxel)
    const int Nblk = blockIdx.y * 64;          // WG col base (channel)

    const int mr = lane & 15;                  // A row / B col within tile
    const int kq = (lane >> 4) << 1;           // K sub-offset: 0 or 2

    v8f acc = {};
    for (int k0 = 0; k0 < NPIX; k0 += KC) {
        // ---- stage A chunk: 32 rows x 64 floats = 512 b128, 2 per thread ----
#pragma unroll
        for (int j = 0; j < 2; ++j) {
            int q = tid + 256 * j;
            int r = q >> 4, c16 = q & 15;
            async_copy_b128(W + (size_t)(Mblk + r) * NPIX + k0 + c16 * 4,
                            &As[r * A_STRIDE + c16 * 4]);
        }
        // ---- stage B chunk: 64 rows x 64 floats = 1024 b128, 4 per thread ----
#pragma unroll
        for (int j = 0; j < 4; ++j) {
            int q = tid + 256 * j;
            int r = q >> 4, c16 = q & 15;
            async_copy_b128(featT + (size_t)(k0 + r) * C_CH + Nblk + c16 * 4,
                            &Bs[r * B_STRIDE + c16 * 4]);
        }
        wait_async_copies();
        __syncthreads();

        // ---- 16 WMMAs over this K-chunk, fragments from LDS ----
#pragma unroll
        for (int kk = 0; kk < KC; kk += 4) {
            v2f a = *(const v2f*)&As[(tm * 16 + mr) * A_STRIDE + kk + kq];
            v2f b;
            b.x = Bs[(kk + kq) * B_STRIDE + tn * 16 + mr];
            b.y = Bs[(kk + kq + 1) * B_STRIDE + tn * 16 + mr];
            acc = __builtin_amdgcn_wmma_f32_16x16x4_f32(
                /*neg_a=*/false, a, /*neg_b=*/false, b,
                /*c_mod=*/(short)0, acc, /*reuse_a=*/false, /*reuse_b=*/false);
        }
        __syncthreads();
    }

    // C/D layout: VGPR r, lanes 0-15 -> (M=r, N=lane), lanes 16-31 -> (M=r+8).
    const int n  = Nblk + tn * 16 + (lane & 15);
    const int mh = (lane >> 4) << 3;           // 0 or 8
#pragma unroll
    for (int r = 0; r < 8; ++r) {
        int vm = Mblk + tm * 16 + mh + r;      // local voxel id
        int vx = BX0 + (vm >> 5);              // / BNY
        int vy = BY0 + (vm & (BNY - 1));
        out[(size_t)n * (NX0 * NX1) + vx * NX1 + vy] = acc[r];
    }
}

// ---------------------------------------------------------------------------
extern "C" void kernel_launch(void* const* d_in, const int* in_sizes, int n_in,
                              void* d_out, int out_size, void* d_ws, size_t ws_size,
                              hipStream_t stream) {
    const float* feat  = (const float*)d_in[0];   // (1,512,24,40)
    const float* depth = (const float*)d_in[1];   // (1,80,24,40)
    float* out = (float*)d_out;                   // (1,512,200,200)

    float* W     = (float*)d_ws;                                  // 1280*960 f32
    float* featT = (float*)((char*)d_ws +
                            (size_t)MVOX * NPIX * sizeof(float)); // 960*512 f32

    // 1) zero output grid + W accumulator (deterministic each launch)
    {
        int n4 = OUT_N / 4;
        lss_zero4<<<(n4 + 255) / 256, 256, 0, stream>>>((float4*)out, n4);
        int w4 = (MVOX * NPIX) / 4;
        lss_zero4<<<(w4 + 255) / 256, 256, 0, stream>>>((float4*)W, w4);
    }
    // 2) transpose features to (P, C)
    {
        int n = NPIX * C_CH;
        lss_transpose<<<(n + 255) / 256, 256, 0, stream>>>(feat, featT);
    }
    // 3) voxelize frustum, accumulate depth weights into W
    lss_build_w<<<(NPTS + 255) / 256, 256, 0, stream>>>(depth, W);

    // 4) WMMA GEMM over the occupied voxel box, scatter into BEV grid
    lss_wmma_gemm<<<dim3(MVOX / 32, C_CH / 64), 256, 0, stream>>>(W, featT, out);
}